// _ObjPairLayer_88313117540567
// MI455X (gfx1250) — compile-verified
//
#include <hip/hip_runtime.h>
#include <hip/hip_bf16.h>
#include <stdint.h>
#include <stddef.h>

// Row = C*W*H floats (512*7*7)
#define ROW_FLOATS 25088
#define ROW_VEC4   (ROW_FLOATS / 4)   // 6272 x 16B packets per row
#define CHUNK_VEC4 2048               // 32 KB LDS staging chunk
#define MAXB       1024

// CPol temporal hints (gfx12-family encoding: TH in bits [2:0])
#define CPOL_RT 0
#define CPOL_NT 1

// ---------------------------------------------------------------------------
// CDNA5 async global<->LDS path (ASYNCcnt). Guarded: falls back to a plain
// VGPR copy if the builtins are absent on this toolchain.
// ---------------------------------------------------------------------------
#if defined(__HIP_DEVICE_COMPILE__) && defined(__gfx1250__) && \
    __has_builtin(__builtin_amdgcn_global_load_async_to_lds_b128) && \
    __has_builtin(__builtin_amdgcn_global_store_async_from_lds_b128)
#define USE_ASYNC_LDS 1
#else
#define USE_ASYNC_LDS 0
#endif

#if USE_ASYNC_LDS
typedef int v4i __attribute__((vector_size(16)));          // 16B packet
typedef __attribute__((address_space(1))) v4i  as1_v4i;    // global int4
typedef __attribute__((address_space(3))) v4i  as3_v4i;    // LDS int4
typedef __attribute__((address_space(1))) char as1_char;
typedef __attribute__((address_space(3))) char as3_char;

__device__ __forceinline__ void wait_async0() {
#if __has_builtin(__builtin_amdgcn_s_wait_asynccnt)
  __builtin_amdgcn_s_wait_asynccnt(0);
#else
  asm volatile("s_wait_asynccnt 0" ::: "memory");
#endif
}
#endif

// ---------------------------------------------------------------------------
// Kernel A: rebuild the pair-index table on device (replaces _pair_indices).
// For each image b: rows [begin+o1, begin+o2, begin+n+pair_counter] with
// (o1,o2) enumerated row-major over triu(n, k=1); begin += n*(n+1)/2.
// table[3*q + j] = source row of output row 3*q + j.
// ---------------------------------------------------------------------------
__global__ void __launch_bounds__(256)
obj_pair_build_table(const int* __restrict__ obj_num, int B, int P,
                     int* __restrict__ table) {
  __shared__ int s_begin[MAXB];
  __shared__ int s_pair[MAXB + 1];
  if (threadIdx.x == 0) {
    int begin = 0, pair = 0;
    int nb = B < MAXB ? B : MAXB;
    for (int b = 0; b < nb; ++b) {
      s_begin[b] = begin;
      s_pair[b]  = pair;
      int n = obj_num[b];
      begin += n * (n + 1) / 2;
      if (n > 1) pair += n * (n - 1) / 2;
    }
    s_pair[nb] = pair;
  }
  __syncthreads();
  int nb = B < MAXB ? B : MAXB;
  for (int q = (int)threadIdx.x; q < P; q += (int)blockDim.x) {
    // locate image
    int b = 0;
    while (b + 1 < nb && q >= s_pair[b + 1]) ++b;
    int p = q - s_pair[b];
    int n = obj_num[b];
    // decode row-major triu pair index p -> (o1, o2)
    int o1 = 0, rem = p;
    while (rem >= n - 1 - o1) { rem -= (n - 1 - o1); ++o1; }
    int o2 = o1 + 1 + rem;
    int begin = s_begin[b];
    table[3 * q + 0] = begin + o1;
    table[3 * q + 1] = begin + o2;
    table[3 * q + 2] = begin + n + p;  // union feat: sequential pair counter
  }
}

// ---------------------------------------------------------------------------
// Kernel B: one block per output row; stream 100 KB row through LDS using
// CDNA5 async B128 DMA ops. Each wave stores exactly the LDS region it
// loaded, so per-wave s_wait_asynccnt is the only sync needed.
// Loads: RT (input rows have ~15x reuse and mostly fit the 192 MB L2).
// Stores: NT (output is write-once; keep it from thrashing L2).
// ---------------------------------------------------------------------------
__global__ void __launch_bounds__(256)
obj_pair_gather_rows(const float* __restrict__ src,
                     const int* __restrict__ table,
                     float* __restrict__ out, int nrows) {
  int r = (int)blockIdx.x;
  if (r >= nrows) return;
  int srow = table[r];
  const char* sp = (const char*)(src + (size_t)srow * ROW_FLOATS);
  char*       dp = (char*)(out + (size_t)r * ROW_FLOATS);

#if USE_ASYNC_LDS
  __shared__ __align__(16) char smem[CHUNK_VEC4 * 16];
  as3_char* lds = (as3_char*)smem;
  as1_char* gs  = (as1_char*)sp;
  as1_char* gd  = (as1_char*)dp;

  for (int base = 0; base < ROW_VEC4; base += CHUNK_VEC4) {
    int cnt = ROW_VEC4 - base;
    if (cnt > CHUNK_VEC4) cnt = CHUNK_VEC4;

    // LDS buffer reuse: previous chunk's async stores must have read it.
    wait_async0();
    for (int j = (int)threadIdx.x; j < cnt; j += 256) {
      __builtin_amdgcn_global_load_async_to_lds_b128(
          (as1_v4i*)(gs + ((size_t)(base + j) << 4)),
          (as3_v4i*)(lds + ((size_t)j << 4)), 0, CPOL_RT);
    }
    // Loads landed in LDS (per-wave ASYNCcnt; loads complete in order).
    wait_async0();
    for (int j = (int)threadIdx.x; j < cnt; j += 256) {
      __builtin_amdgcn_global_store_async_from_lds_b128(
          (as1_v4i*)(gd + ((size_t)(base + j) << 4)),
          (as3_v4i*)(lds + ((size_t)j << 4)), 0, CPOL_NT);
    }
  }
  // S_ENDPGM performs an implicit wait-idle, draining outstanding stores.
#else
  const float4* s4 = (const float4*)sp;
  float4*       d4 = (float4*)dp;
  for (int j = (int)threadIdx.x; j < ROW_VEC4; j += 256)
    d4[j] = s4[j];
#endif
}

// ---------------------------------------------------------------------------
extern "C" void kernel_launch(void* const* d_in, const int* in_sizes, int n_in,
                              void* d_out, int out_size, void* d_ws, size_t ws_size,
                              hipStream_t stream) {
  const float* feats   = (const float*)d_in[0];
  const int*   obj_num = (const int*)d_in[2];
  int B = in_sizes[2];                       // obj_num has one entry per image

  int nrows = out_size / ROW_FLOATS;         // total output rows (= 3 * P)
  int P     = nrows / 3;

  int* table = (int*)d_ws;                   // nrows ints of scratch

  obj_pair_build_table<<<1, 256, 0, stream>>>(obj_num, B, P, table);
  obj_pair_gather_rows<<<nrows, 256, 0, stream>>>(feats, table,
                                                  (float*)d_out, nrows);
}